// PseudoTransConvLayer_47717086658594
// MI455X (gfx1250) — compile-verified
//
#include <hip/hip_runtime.h>
#include <math.h>

// ---------------- problem constants ----------------
#define NR      30000        // N rows
#define HEADS   8
#define IND     257          // IN_DIM == OUT_DIM
#define OT1     17           // OT+1 (s-part width)
#define HD      240          // OS (h-part width)
#define SP      32           // padded s-part
#define DP      272          // padded full row: 32 (s) + 240 (h)
#define KP      260          // padded K for projections
#define EPSF    1e-8f
#define MAXN    1000000.0f
#define NT      1875         // NR/16 n-tiles

#define WT_ELEMS  ((size_t)3 * HEADS * KP * DP)   // 1,697,280
#define WHT_ELEMS ((size_t)(HEADS * HD) * HD)     // 460,800

typedef __attribute__((ext_vector_type(2))) float v2f;
typedef __attribute__((ext_vector_type(8))) float v8f;

// fp32 WMMA: D(16x16) += A(16x4) * B(4x16), full fp32 accumulate
#define WMMA_F32(a, b, c) \
  __builtin_amdgcn_wmma_f32_16x16x4_f32(false, (a), false, (b), (short)0, (c), false, false)

__device__ __forceinline__ float elu1(float x) {
  return x > 0.0f ? x + 1.0f : expf(x);
}

// =====================================================================
// Kernel 0: pad + transpose weights into workspace, K-PAIR interleaved:
//  WT2[z][h][k/2][c][p]  (p = k&1): one b64 per B fragment in k_proj
//  WhT2[k/2][j][p]       : one b64 per B fragment in k_hfinal
// Zeros for k>=257 and c in [17,32).
// =====================================================================
__global__ void k_prep(const float* __restrict__ Wq, const float* __restrict__ Wk,
                       const float* __restrict__ Wv, const float* __restrict__ Wh,
                       float* __restrict__ WT, float* __restrict__ WhT)
{
  const size_t i = (size_t)blockIdx.x * blockDim.x + threadIdx.x;
  if (i < WT_ELEMS) {
    const int zh  = (int)(i / (KP * DP));
    const int rem = (int)(i % (KP * DP));
    const int kp  = rem / (DP * 2);
    const int r2  = rem % (DP * 2);
    const int c   = r2 >> 1, p = r2 & 1;
    const int k   = kp * 2 + p;
    const int z = zh >> 3, h = zh & 7;
    const int o = (c < SP) ? c : (c - (SP - OT1));
    const bool valid = ((c < OT1) || (c >= SP)) && (k < IND);
    const float* W = (z == 0) ? Wq : (z == 1 ? Wk : Wv);
    WT[i] = valid ? W[((size_t)h * IND + o) * IND + k] : 0.0f;
  } else if (i < WT_ELEMS + WHT_ELEMS) {
    const size_t i2 = i - WT_ELEMS;
    const int kp = (int)(i2 / (HD * 2));
    const int r2 = (int)(i2 % (HD * 2));
    const int j  = r2 >> 1, p = r2 & 1;
    const int k  = kp * 2 + p;
    WhT[i2] = Wh[(size_t)j * (HEADS * HD) + k];
  }
}

// =====================================================================
// Kernel 1: head-batched projection  Y[n,h,c] = sum_k X[n,k] * W[z,h,o(c),k]
// grid (NT, 8, 3): z=0 q(query) z=1 k(source) z=2 v(source)
// =====================================================================
__global__ void k_proj(const float* __restrict__ Xq, const float* __restrict__ Xs,
                       const float* __restrict__ WT,
                       float* __restrict__ Yq, float* __restrict__ Yk, float* __restrict__ Yv)
{
  const int n0 = blockIdx.x * 16;
  const int h  = blockIdx.y;
  const int z  = blockIdx.z;
  const float* X  = (z == 0) ? Xq : Xs;
  float*       Y  = (z == 0) ? Yq : (z == 1 ? Yk : Yv);
  const float* BT = WT + (size_t)(z * HEADS + h) * (KP * DP);   // pair-interleaved

  __shared__ float lx[16][KP];                      // 16 rows x K (zero padded)
  for (int idx = threadIdx.x; idx < 16 * KP; idx += blockDim.x) {
    int rr = idx / KP, kk = idx % KP;
    lx[rr][kk] = (kk < IND) ? X[(size_t)(n0 + rr) * IND + kk] : 0.0f;
  }
  __syncthreads();

  const int lane = threadIdx.x & 31;
  const int wave = threadIdx.x >> 5;
  const int r    = lane & 15;
  const int hs   = lane >> 4;
  const int kkb  = hs * 2;

  for (int t = wave; t < 17; t += 8) {              // 17 output col-tiles of 16
    const int c = t * 16 + r;
    const v2f*   bp = (const v2f*)BT + (size_t)(kkb >> 1) * DP + c;  // running B ptr
    const float* ap = &lx[r][kkb];                                   // running A ptr
    v8f acc = {};
    for (int k0 = 0; k0 < KP; k0 += 4) {
      v2f a;
      a.x = ap[0];
      a.y = ap[1];
      const v2f b = *bp;
      acc = WMMA_F32(a, b, acc);
      ap += 4;
      bp += 2 * DP;
    }
#pragma unroll
    for (int i = 0; i < 8; i++) {
      const int row = i + 8 * hs;
      Y[((size_t)(n0 + row) * HEADS + h) * DP + c] = acc[i];
    }
  }
}

// =====================================================================
// Kernel 2: per-(n,h) transform: normalize s-part, elu+1 for q/k, zero pads
// grid NR, block 256 (one wave per head)
// =====================================================================
__global__ void k_xform(float* __restrict__ Yq, float* __restrict__ Yk, float* __restrict__ Yv)
{
  const int n    = blockIdx.x;
  const int h    = threadIdx.x >> 5;
  const int lane = threadIdx.x & 31;

  for (int z = 0; z < 3; z++) {
    float* Y   = (z == 0) ? Yq : (z == 1 ? Yk : Yv);
    float* row = Y + ((size_t)n * HEADS + h) * DP;
    float sv = (lane < OT1) ? row[lane] : 0.0f;
    float ss = sv * sv;
#pragma unroll
    for (int off = 16; off > 0; off >>= 1) ss += __shfl_xor(ss, off, 32);
    const float scale = 1.0f / (sqrtf(ss) + EPSF);  // SQRT_BETA == 1
    const float s = sv * scale;
    if (z < 2) {
      row[lane] = (lane < OT1) ? elu1(s) : 0.0f;    // also zeros pads [17,32)
      for (int j = SP + lane; j < DP; j += 32) row[j] = elu1(row[j]);
    } else {
      row[lane] = (lane < OT1) ? s : 0.0f;          // v: normalized s, raw h
    }
  }
}

// =====================================================================
// Kernel 3: ktv[h,m,d] = sum_n phiK[n,h,m] * V[n,h,d]  (WMMA, K = n)
// grid (229 tiles, 8 heads, 15 n-chunks of 2000), block = 1 wave
// tiles 0..3 -> ktvS (32x32), tiles 4..228 -> ktvH (240x240)
// Output stored M-PAIR interleaved: ktv2[(m/2)*ld + d][p = m&1]
// =====================================================================
__global__ void k_ktv(const float* __restrict__ phiK, const float* __restrict__ V,
                      float* __restrict__ ktvS, float* __restrict__ ktvH)
{
  const int tile = blockIdx.x;
  const int h    = blockIdx.y;
  const int lane = threadIdx.x & 31;
  const int r = lane & 15, hs = lane >> 4, kkb = hs * 2;

  int m0, d0, koff, ldo;  float* out;
  if (tile < 4) { m0 = (tile >> 1) * 16; d0 = (tile & 1) * 16; koff = 0;  out = ktvS + h * SP * SP; ldo = SP; }
  else { const int t = tile - 4; m0 = (t / 15) * 16; d0 = (t % 15) * 16; koff = SP; out = ktvH + h * HD * HD; ldo = HD; }

  const size_t stride = (size_t)HEADS * DP;
  const int nbeg = blockIdx.z * 2000;
  const float* pa = phiK + (size_t)h * DP + koff + (size_t)(nbeg + kkb) * stride + m0 + r;
  const float* pb = V    + (size_t)h * DP + koff + (size_t)(nbeg + kkb) * stride + d0 + r;

  v8f acc = {};
  for (int it = 0; it < 500; it++) {                // 2000 rows, 4 per WMMA
    v2f a, b;
    a.x = pa[0];
    a.y = pa[stride];
    b.x = pb[0];
    b.y = pb[stride];
    acc = WMMA_F32(a, b, acc);
    pa += 4 * stride;
    pb += 4 * stride;
  }
#pragma unroll
  for (int i = 0; i < 8; i++) {
    const int m = m0 + i + 8 * hs;
    const int d = d0 + r;
    atomicAdd(&out[((size_t)(m >> 1) * ldo + d) * 2 + (m & 1)], acc[i]);
  }
}

// =====================================================================
// Kernel 3b: sk[h,d] = sum_n phiK[n,h,d]
// grid (34, 30), block 64  (H*DP = 2176 cols, 30 n-chunks of 1000)
// =====================================================================
__global__ void k_sk(const float* __restrict__ phiK, float* __restrict__ skS, float* __restrict__ skH)
{
  const int col = blockIdx.x * 64 + threadIdx.x;    // 0..2175
  const int h = col / DP, d = col % DP;
  const int nbeg = blockIdx.y * 1000, nend = nbeg + 1000;
  const float* p = phiK + (size_t)h * DP + d;
  float s = 0.0f;
  for (int n = nbeg; n < nend; n++) s += p[(size_t)n * HEADS * DP];
  if (d < SP) atomicAdd(&skS[h * SP + d], s);
  else        atomicAdd(&skH[h * HD + (d - SP)], s);
}

// =====================================================================
// Kernel 4: num/den -> a (s path, Abuf (N,H,32)) and b (h path, Bbuf (N,H,240))
// grid (NT, 8), block 256.  ktv read via pair-interleaved b64 loads.
// =====================================================================
__global__ void k_attn(const float* __restrict__ phiQ, const float* __restrict__ ktvS,
                       const float* __restrict__ ktvH, const float* __restrict__ skS,
                       const float* __restrict__ skH,
                       float* __restrict__ Abuf, float* __restrict__ Bbuf)
{
  const int n0 = blockIdx.x * 16;
  const int h  = blockIdx.y;
  __shared__ float lq[16][DP + 4];
  __shared__ float denS[16], denH[16];

  for (int idx = threadIdx.x; idx < 16 * DP; idx += blockDim.x) {
    int rr = idx / DP, kk = idx % DP;
    lq[rr][kk] = phiQ[((size_t)(n0 + rr) * HEADS + h) * DP + kk];
  }
  __syncthreads();
  if (threadIdx.x < 16) {
    float ds = 0.0f;
    for (int d = 0; d < SP; d++) ds += lq[threadIdx.x][d] * skS[h * SP + d];
    denS[threadIdx.x] = ds + 1e-6f;
  } else if (threadIdx.x < 32) {
    const int rr = threadIdx.x - 16;
    float dh = 0.0f;
    for (int d = 0; d < HD; d++) dh += lq[rr][SP + d] * skH[h * HD + d];
    denH[rr] = dh + 1e-8f;
  }
  __syncthreads();

  const int lane = threadIdx.x & 31, wave = threadIdx.x >> 5;
  const int r = lane & 15, hs = lane >> 4, kkb = hs * 2;

  for (int t = wave; t < 17; t += 8) {
    v8f acc = {};
    if (t < 2) {
      const int d0 = t * 16;
      const v2f*   bp = (const v2f*)(ktvS + h * SP * SP) + (size_t)(kkb >> 1) * SP + d0 + r;
      const float* ap = &lq[r][kkb];
      for (int k0 = 0; k0 < SP; k0 += 4) {
        v2f a;
        a.x = ap[0];  a.y = ap[1];
        const v2f b = *bp;
        acc = WMMA_F32(a, b, acc);
        ap += 4;
        bp += 2 * SP;
      }
#pragma unroll
      for (int i = 0; i < 8; i++) {
        const int row = i + 8 * hs;
        Abuf[((size_t)(n0 + row) * HEADS + h) * SP + d0 + r] = acc[i] / denS[row];
      }
    } else {
      const int d0 = (t - 2) * 16;
      const v2f*   bp = (const v2f*)(ktvH + h * HD * HD) + (size_t)(kkb >> 1) * HD + d0 + r;
      const float* ap = &lq[r][SP + kkb];
      for (int k0 = 0; k0 < HD; k0 += 4) {
        v2f a;
        a.x = ap[0];  a.y = ap[1];
        const v2f b = *bp;
        acc = WMMA_F32(a, b, acc);
        ap += 4;
        bp += 2 * HD;
      }
#pragma unroll
      for (int i = 0; i < 8; i++) {
        const int row = i + 8 * hs;
        Bbuf[((size_t)(n0 + row) * HEADS + h) * HD + d0 + r] = acc[i] / denH[row];
      }
    }
  }
}

// =====================================================================
// Kernel 5: s-path output: att_s[n] from a(N,H*17) @ Ws^T + clamp/l2norm
// grid NR, block 32 (one wave per row; tiny K=136)
// =====================================================================
__global__ void k_sfinal(const float* __restrict__ Abuf, const float* __restrict__ Ws,
                         float* __restrict__ attS)
{
  const int n = blockIdx.x;
  const int lane = threadIdx.x;
  float acc = 0.0f;
  if (lane < OT1) {
    const float* w = Ws + (size_t)lane * (HEADS * OT1);   // Ws (17, 136)
    const float* a = Abuf + (size_t)n * HEADS * SP;
    for (int h = 0; h < HEADS; h++)
      for (int m = 0; m < OT1; m++)
        acc += a[h * SP + m] * w[h * OT1 + m];
  }
  float ss = (lane < OT1) ? acc * acc : 0.0f;
#pragma unroll
  for (int off = 16; off > 0; off >>= 1) ss += __shfl_xor(ss, off, 32);
  const float an = sqrtf(ss) + EPSF;
  const float a_ = (an > MAXN) ? acc / MAXN : acc;        // an_c = MAXN when masked
  float s2 = (lane < OT1) ? a_ * a_ : 0.0f;
#pragma unroll
  for (int off = 16; off > 0; off >>= 1) s2 += __shfl_xor(s2, off, 32);
  const float na = fmaxf(sqrtf(s2), 1e-12f);
  if (lane < SP) attS[(size_t)n * SP + lane] = (lane < OT1) ? (a_ / na) : 0.0f;
}

// =====================================================================
// Kernel 6: h-path GEMM b(N,1920) @ Wh^T + epilogue + final concat
// grid NT, block 256.  WhT2 pair-interleaved -> b64 B loads.
// =====================================================================
__global__ void k_hfinal(const float* __restrict__ Bbuf, const float* __restrict__ WhT,
                         const float* __restrict__ attS, float* __restrict__ outp)
{
  const int n0 = blockIdx.x * 16;
  __shared__ float lb[16][248];       // K-chunk of b rows
  __shared__ float lo[16][HD];        // projected b
  __shared__ float sBT[16], sF[16];

  const int lane = threadIdx.x & 31, wave = threadIdx.x >> 5;
  const int r = lane & 15, hs = lane >> 4, kkb = hs * 2;
  const int t0 = wave, t1 = wave + 8;
  const int j0 = t0 * 16 + r, j1 = t1 * 16 + r;

  v8f acc0 = {}, acc1 = {};
  for (int kc = 0; kc < 8; kc++) {    // K = 1920 in chunks of 240
    __syncthreads();
    for (int idx = threadIdx.x; idx < 16 * HD; idx += blockDim.x) {
      int rr = idx / HD, kk = idx % HD;
      lb[rr][kk] = Bbuf[(size_t)(n0 + rr) * (HEADS * HD) + kc * HD + kk];
    }
    __syncthreads();
    const v2f*   b0 = (const v2f*)WhT + ((size_t)kc * (HD / 2) + (kkb >> 1)) * HD + j0;
    const v2f*   b1 = (const v2f*)WhT + ((size_t)kc * (HD / 2) + (kkb >> 1)) * HD + j1;
    const float* ap = &lb[r][kkb];
    for (int k0 = 0; k0 < HD; k0 += 4) {
      v2f a;
      a.x = ap[0];  a.y = ap[1];
      {
        const v2f b = *b0;
        acc0 = WMMA_F32(a, b, acc0);
      }
      if (t1 < 15) {
        const v2f b = *b1;
        acc1 = WMMA_F32(a, b, acc1);
      }
      ap += 4;
      b0 += 2 * HD;
      b1 += 2 * HD;
    }
  }
  __syncthreads();
#pragma unroll
  for (int i = 0; i < 8; i++) {
    const int row = i + 8 * hs;
    lo[row][j0] = acc0[i];
    if (t1 < 15) lo[row][j1] = acc1[i];
  }
  __syncthreads();
  if (threadIdx.x < 16) {
    const int rr = threadIdx.x;
    float ss = 0.0f;
    for (int j = 0; j < HD; j++) { float v = lo[rr][j]; ss += v * v; }
    const float nb  = sqrtf(ss);
    const float bn  = nb + EPSF;
    const float bnc = fminf(bn, MAXN);
    sBT[rr] = sqrtf(bnc * bnc + 1.0f);                       // -BETA = 1
    sF[rr]  = (bn > MAXN) ? (MAXN / fmaxf(nb, 1e-12f)) : 1.0f;
  }
  __syncthreads();
  for (int idx = threadIdx.x; idx < 16 * IND; idx += blockDim.x) {
    const int rr = idx / IND, j = idx % IND;
    const int n = n0 + rr;
    float val;
    if (j < OT1) val = sBT[rr] * attS[(size_t)n * SP + j];
    else         val = sF[rr] * lo[rr][j - OT1];
    outp[(size_t)n * IND + j] = val;
  }
}

// =====================================================================
extern "C" void kernel_launch(void* const* d_in, const int* in_sizes, int n_in,
                              void* d_out, int out_size, void* d_ws, size_t ws_size,
                              hipStream_t stream)
{
  const float* Xq = (const float*)d_in[0];
  const float* Xs = (const float*)d_in[1];
  const float* Wq = (const float*)d_in[2];
  const float* Wk = (const float*)d_in[3];
  const float* Wv = (const float*)d_in[4];
  const float* Ws = (const float*)d_in[5];
  const float* Wh = (const float*)d_in[6];
  float* outp = (float*)d_out;

  const size_t big = (size_t)NR * HEADS * DP;       // 65,280,000 floats
  float* ws   = (float*)d_ws;
  float* phiQ = ws;
  float* phiK = phiQ + big;
  float* Vb   = phiK + big;
  float* ktvS = Vb + big;
  float* ktvH = ktvS + HEADS * SP * SP;
  float* skS  = ktvH + HEADS * HD * HD;
  float* skH  = skS + HEADS * SP;
  float* WT   = skH + HEADS * HD;
  float* WhT  = WT + WT_ELEMS;
  // overlays (safe by kernel ordering)
  float* Bbuf = phiK;                               // (N, 1920), used from kernel 4 on
  float* attS = phiK + (size_t)NR * HEADS * HD;     // (N, 32)
  float* Abuf = Vb;                                 // (N, H, 32), used from kernel 4 on

  const size_t zbytes = (size_t)(HEADS * SP * SP + HEADS * HD * HD + HEADS * SP + HEADS * HD) * sizeof(float);
  hipMemsetAsync(ktvS, 0, zbytes, stream);

  const int prep_total = (int)(WT_ELEMS + WHT_ELEMS);
  k_prep  <<<dim3((prep_total + 255) / 256), 256, 0, stream>>>(Wq, Wk, Wv, Wh, WT, WhT);
  k_proj  <<<dim3(NT, HEADS, 3), 256, 0, stream>>>(Xq, Xs, WT, phiQ, phiK, Vb);
  k_xform <<<dim3(NR), 256, 0, stream>>>(phiQ, phiK, Vb);
  k_ktv   <<<dim3(229, HEADS, 15), 32, 0, stream>>>(phiK, Vb, ktvS, ktvH);
  k_sk    <<<dim3(34, 30), 64, 0, stream>>>(phiK, skS, skH);
  k_attn  <<<dim3(NT, HEADS), 256, 0, stream>>>(phiQ, ktvS, ktvH, skS, skH, Abuf, Bbuf);
  k_sfinal<<<dim3(NR), 32, 0, stream>>>(Abuf, Ws, attS);
  k_hfinal<<<dim3(NT), 256, 0, stream>>>(Bbuf, WhT, attS, outp);
}